// MessagePassing_7507602833984
// MI455X (gfx1250) — compile-verified
//
#include <hip/hip_runtime.h>

typedef __attribute__((ext_vector_type(2))) float v2f;
typedef __attribute__((ext_vector_type(8))) float v8f;
typedef __attribute__((ext_vector_type(4))) unsigned u32x4;
typedef __attribute__((ext_vector_type(8))) int     i32x8;
typedef __attribute__((ext_vector_type(4))) int     i32x4;

#define NV    50000
#define DDIM  512
#define HDIM  512
#define NL    4
#define NE    40000

// ---------------- Kernel 0: zero the output accumulator ----------------
__global__ __launch_bounds__(256) void zero_f32(float* __restrict__ p, long long n4) {
    long long i = (long long)blockIdx.x * blockDim.x + threadIdx.x;
    long long stride = (long long)gridDim.x * blockDim.x;
    float4* p4 = (float4*)p;
    for (; i < n4; i += stride) p4[i] = make_float4(0.f, 0.f, 0.f, 0.f);
}

__device__ __forceinline__ unsigned rfl(unsigned x) {
    return (unsigned)__builtin_amdgcn_readfirstlane((int)x);
}

// ---------------- Kernel 1: TDM gather -> WMMA GEMM -> atomic scatter ----------------
__global__ __launch_bounds__(256) void mp_gemm_scatter(
    const float* __restrict__ emb,   // [NV, DDIM]
    const int*   __restrict__ adj,   // [NL, NE, 2]
    const float* __restrict__ W,     // [NL, DDIM, HDIM]
    float*       __restrict__ out)   // [NV, HDIM] accumulator
{
    __shared__ float Atile[16][DDIM];   // 32 KB gathered source rows (TDM destination)
    __shared__ int   tgts[16];

    const int tilesPerLayer = NE / 16;              // 2500
    const int l  = blockIdx.x / tilesPerLayer;      // edge type
    const int e0 = (blockIdx.x % tilesPerLayer) * 16;

    const int tid = threadIdx.x;
    const int* ap = adj + ((long long)l * NE + e0) * 2;   // 16 (src,tgt) pairs, uniform address

    // Wave 1 (lanes 32..47): stage target indices for the scatter epilogue.
    if (tid >= 32 && tid < 48) {
        tgts[tid - 32] = ap[(tid - 32) * 2 + 1];
    }

    // Wave 0: build the TDM gather descriptor (all values wave-uniform -> SGPRs)
    // and issue TENSOR_LOAD_TO_LDS: 16 rows x 512 f32 gathered by 16-bit row index.
    if (tid < 32) {
        const unsigned long long gaddr = (unsigned long long)(uintptr_t)emb;
        const unsigned lds_addr = (unsigned)(uintptr_t)&Atile[0][0]; // low 32 bits of LDS aperture = LDS byte offset

        // ---- D# group 0: count=1, gather_mode=1 (16-bit indices), lds_addr, global_addr, type=2
        u32x4 g0;
        g0.x = rfl(0x80000001u);                       // count=1 | gather_mode<<31 (index_size=0 -> 16-bit)
        g0.y = rfl(lds_addr);                          // lds_addr [63:32]
        g0.z = rfl((unsigned)(gaddr & 0xFFFFFFFFu));   // global_addr [95:64]
        g0.w = rfl((unsigned)((gaddr >> 32) & 0x01FFFFFFu) | (2u << 30)); // global_addr[56:32] | type=2

        // ---- D# group 1: data_size=4B; tensor_dim0=512; tensor_dim1=NV; tile_dim0=512;
        //                  tile_dim1=16 valid indices; tensor_dim0_stride=512
        i32x8 g1;
        g1[0] = (int)rfl(2u << 16);                    // data_size=2 (4 bytes)
        g1[1] = (int)rfl((unsigned)(DDIM) << 16);      // tensor_dim0[15:0] in bits 63:48
        g1[2] = (int)rfl((unsigned)(NV & 0xFFFF) << 16);  // tensor_dim0 hi=0 | tensor_dim1[15:0]
        g1[3] = (int)rfl((unsigned)(DDIM) << 16);      // tensor_dim1 hi=0 | tile_dim0=512
        g1[4] = (int)rfl(16u);                         // tile_dim1 = 16 valid indices (tile_dim2 ignored)
        g1[5] = (int)rfl((unsigned)DDIM);              // tensor_dim0_stride[31:0] = 512 elements
        g1[6] = (int)rfl(0u);                          // stride hi | dim1_stride (ignored in gather mode)
        g1[7] = (int)rfl(0u);

        // ---- D# groups 2/3: sixteen 16-bit row indices = source node IDs (< 65536)
        i32x4 g2, g3;
        #pragma unroll
        for (int j = 0; j < 4; ++j) {
            unsigned lo = (unsigned)ap[4 * j]     & 0xFFFFu;       // row_index 2j
            unsigned hi = (unsigned)ap[4 * j + 2] & 0xFFFFu;       // row_index 2j+1
            g2[j] = (int)rfl(lo | (hi << 16));
            lo = (unsigned)ap[16 + 4 * j]     & 0xFFFFu;           // row_index 8+2j
            hi = (unsigned)ap[16 + 4 * j + 2] & 0xFFFFu;           // row_index 8+2j+1
            g3[j] = (int)rfl(lo | (hi << 16));
        }

        // 6-arg toolchain variant: trailing int32x8 (unused here) + cpol.
        i32x8 gpad = {0, 0, 0, 0, 0, 0, 0, 0};
        __builtin_amdgcn_tensor_load_to_lds(g0, g1, g2, g3, gpad, 0);
        __builtin_amdgcn_s_wait_tensorcnt(0);
    }
    __syncthreads();   // publish TDM completion + tgts to all 8 waves

    const int wave  = tid >> 5;              // 0..7 -> owns 64 output columns
    const int lane  = tid & 31;
    const int mrow  = lane & 15;             // M row (A) / N column (B,C)
    const int khalf = (lane < 16) ? 0 : 2;   // f32 A/B fragment K-half per ISA layout

    const float* Wl = W + (long long)l * DDIM * HDIM;
    const int nbase = wave * 64;

    v8f acc0 = {}, acc1 = {}, acc2 = {}, acc3 = {};

    #pragma unroll 2
    for (int k = 0; k < DDIM; k += 4) {
        // A fragment: lanes 0-15 -> A[m][k..k+1]; lanes 16-31 -> A[m][k+2..k+3]
        v2f a = *(const v2f*)(&Atile[mrow][k + khalf]);

        const float* wrow0 = Wl + (long long)(k + khalf)     * HDIM;
        const float* wrow1 = Wl + (long long)(k + khalf + 1) * HDIM;

        {   v2f b; b.x = wrow0[nbase +  0 + mrow]; b.y = wrow1[nbase +  0 + mrow];
            acc0 = __builtin_amdgcn_wmma_f32_16x16x4_f32(false, a, false, b, (short)0, acc0, false, false); }
        {   v2f b; b.x = wrow0[nbase + 16 + mrow]; b.y = wrow1[nbase + 16 + mrow];
            acc1 = __builtin_amdgcn_wmma_f32_16x16x4_f32(false, a, false, b, (short)0, acc1, false, false); }
        {   v2f b; b.x = wrow0[nbase + 32 + mrow]; b.y = wrow1[nbase + 32 + mrow];
            acc2 = __builtin_amdgcn_wmma_f32_16x16x4_f32(false, a, false, b, (short)0, acc2, false, false); }
        {   v2f b; b.x = wrow0[nbase + 48 + mrow]; b.y = wrow1[nbase + 48 + mrow];
            acc3 = __builtin_amdgcn_wmma_f32_16x16x4_f32(false, a, false, b, (short)0, acc3, false, false); }
    }

    // Scatter: C layout -> VGPR r holds M=r (lanes 0-15) and M=r+8 (lanes 16-31),
    // N = nbase + t*16 + mrow. Compute the 8 row base addresses once; the
    // per-N-tile step (t*64 bytes) folds into the atomic's immediate offset.
    const int moff = (lane < 16) ? 0 : 8;
    float* baseaddr[8];
    #pragma unroll
    for (int r = 0; r < 8; ++r)
        baseaddr[r] = out + (long long)tgts[r + moff] * HDIM + nbase + mrow;

    #pragma unroll
    for (int t = 0; t < 4; ++t) {
        const v8f acc = (t == 0) ? acc0 : (t == 1) ? acc1 : (t == 2) ? acc2 : acc3;
        #pragma unroll
        for (int r = 0; r < 8; ++r) {
            (void)__hip_atomic_fetch_add(baseaddr[r] + t * 16, acc[r],
                                         __ATOMIC_RELAXED, __HIP_MEMORY_SCOPE_AGENT);
        }
    }
}

// ---------------- Kernel 2: ReLU after aggregation ----------------
__global__ __launch_bounds__(256) void relu_f32(float* __restrict__ p, long long n4) {
    long long i = (long long)blockIdx.x * blockDim.x + threadIdx.x;
    long long stride = (long long)gridDim.x * blockDim.x;
    float4* p4 = (float4*)p;
    for (; i < n4; i += stride) {
        float4 v = p4[i];
        v.x = fmaxf(v.x, 0.f); v.y = fmaxf(v.y, 0.f);
        v.z = fmaxf(v.z, 0.f); v.w = fmaxf(v.w, 0.f);
        p4[i] = v;
    }
}

extern "C" void kernel_launch(void* const* d_in, const int* in_sizes, int n_in,
                              void* d_out, int out_size, void* d_ws, size_t ws_size,
                              hipStream_t stream) {
    (void)in_sizes; (void)n_in; (void)d_ws; (void)ws_size; (void)out_size;

    const float* emb = (const float*)d_in[0];   // [NV, DDIM] f32
    const int*   adj = (const int*)d_in[1];     // [NL, NE, 2] i32
    const float* W   = (const float*)d_in[2];   // [NL, DDIM, HDIM] f32
    float*       out = (float*)d_out;           // [NV, HDIM] f32

    const long long n4 = (long long)NV * HDIM / 4;   // float4 count
    const int zblocks = (int)((n4 + 255) / 256);

    zero_f32<<<zblocks, 256, 0, stream>>>(out, n4);

    const int gemmBlocks = NL * (NE / 16);           // 10000 tiles
    mp_gemm_scatter<<<gemmBlocks, 256, 0, stream>>>(emb, adj, W, out);

    relu_f32<<<zblocks, 256, 0, stream>>>(out, n4);
}